// WavCodec_39814346833937
// MI455X (gfx1250) — compile-verified
//
#include <hip/hip_runtime.h>
#include <hip/hip_bf16.h>
#include <math.h>

// ---------------------------------------------------------------------------
// CDNA5 (gfx1250) wave32 WMMA types and helpers
// ---------------------------------------------------------------------------
typedef float v2f __attribute__((ext_vector_type(2)));
typedef float v8f __attribute__((ext_vector_type(8)));

__device__ __forceinline__ v8f wmma_f32_16x16x4(v2f a, v2f b, v8f c) {
  // 8 args: (neg_a, A, neg_b, B, c_mod, C, reuse_a, reuse_b)
  return __builtin_amdgcn_wmma_f32_16x16x4_f32(false, a, false, b, (short)0, c,
                                               false, false);
}

__device__ __forceinline__ float gelu_f(float x) {
  // jax.nn.gelu(approximate=False) = 0.5*x*(1+erf(x/sqrt(2)))
  return 0.5f * x * (1.0f + erff(x * 0.70710678118654752440f));
}

__device__ __forceinline__ float sigm_f(float x) {
  return 1.0f / (1.0f + expf(-x));
}

// ---------------------------------------------------------------------------
// Generic GEMM-conv with fp32 WMMA (16x16x4), strength-reduced addressing.
//
//   y[b][o][t*KOUT + kko] = gelu( sum_{r<R} A(o,r) * Bx(b,t,r) + bias[o] )
//
// All operand indices are affine in r with constant steps, kept as 32-bit
// unsigned offsets from uniform base pointers (SGPR base + VGPR offset).
//   A index  = o*wsO + kko*aKko + r            (A pairs always contiguous)
//   Bx index = b*xsB + t*xsT + r*bStepR        (pair offset offB1; BP => ==1)
// One wave computes one 16x16 output tile; K loop = R/4 WMMA ops.
// ---------------------------------------------------------------------------
template <bool BP>
__global__ __launch_bounds__(256) void conv_gemm_wmma(
    const float* __restrict__ x, const float* __restrict__ w,
    const float* __restrict__ bias, float* __restrict__ y,
    int Mtiles,          // O/16
    int nColTiles,       // (B*colT)/16
    int colT,            // t-extent of GEMM columns per batch element
    int R,               // reduction length
    unsigned wsO,        // A row stride
    unsigned aKko,       // A stride for kko term
    unsigned xsB, unsigned xsT,
    unsigned bStepR,     // B element step per unit r (even r)
    unsigned offB1,      // B offset from r to r+1 (==1 when BP)
    unsigned ysB, unsigned ysO, int KOUT)
{
  int gw = (int)(((long)blockIdx.x * blockDim.x + threadIdx.x) >> 5);
  int total = Mtiles * nColTiles;
  if (gw >= total) return;
  int tileM = gw % Mtiles;
  int tileN = gw / Mtiles;
  int kko  = blockIdx.y;
  int lane = threadIdx.x & 31;
  int n    = lane & 15;             // A row / B col index for this lane
  int kb   = (lane >> 4) << 1;      // K base: 0 (lanes 0-15) / 2 (lanes 16-31)

  int col = tileN * 16 + n;
  unsigned cbi = (unsigned)(col / colT);     // batch index
  unsigned cti = (unsigned)(col % colT);     // t index

  unsigned ia = (unsigned)(tileM * 16 + n) * wsO + (unsigned)kko * aKko
              + (unsigned)kb;
  unsigned ib = cbi * xsB + cti * xsT + (unsigned)kb * bStepR;
  unsigned stepB = 4u * bStepR;

  v8f acc = {};
  if (BP) {
#pragma unroll 4
    for (int it = 0; it < R; it += 4) {
      v2f a = *(const v2f*)(w + ia);
      v2f b = *(const v2f*)(x + ib);
      acc = wmma_f32_16x16x4(a, b, acc);
      ia += 4u;
      ib += stepB;
    }
  } else {
    unsigned ib1 = ib + offB1;
#pragma unroll 4
    for (int it = 0; it < R; it += 4) {
      v2f a = *(const v2f*)(w + ia);
      v2f b;
      b.x = x[ib];
      b.y = x[ib1];
      acc = wmma_f32_16x16x4(a, b, acc);
      ia += 4u;
      ib += stepB;
      ib1 += stepB;
    }
  }

  int rbase = (lane & 16) ? 8 : 0;  // C/D layout: VGPR r -> M=r (+8 hi half)
#pragma unroll
  for (int r = 0; r < 8; ++r) {
    unsigned orow = (unsigned)(tileM * 16 + r + rbase);
    float v = gelu_f(acc[r] + bias[orow]);
    y[cbi * ysB + orow * ysO + cti * (unsigned)KOUT + (unsigned)kko] = v;
  }
}

// ---------------------------------------------------------------------------
// Decoder convT weight transpose: w (512,512,2) [c][o][kk] -> wt [kk][o][c]
// (makes the A-operand pair contiguous for WMMA b64 loads)
// ---------------------------------------------------------------------------
__global__ void transpose_w_cok(const float* __restrict__ w,
                                float* __restrict__ wt) {
  int idx = blockIdx.x * blockDim.x + threadIdx.x;   // 524288
  if (idx >= 512 * 512 * 2) return;
  int c = idx >> 10;
  int rest = idx & 1023;
  int o = rest >> 1, kk = rest & 1;
  wt[kk * 262144 + o * 512 + c] = w[idx];
}

// ---------------------------------------------------------------------------
// Encoder layer 0: C_in=1, k=5 (tiny reduction; VALU)
// ---------------------------------------------------------------------------
__global__ void conv0_k5(const float* __restrict__ wav, const float* __restrict__ w,
                         const float* __restrict__ bias, float* __restrict__ y,
                         int Tout)
{
  long idx = (long)blockIdx.x * blockDim.x + threadIdx.x;
  long total = (long)4 * 512 * Tout;
  if (idx >= total) return;
  int t = (int)(idx % Tout);
  int o = (int)((idx / Tout) % 512);
  int b = (int)(idx / ((long)512 * Tout));
  const float* xv = wav + (long)b * Tout * 5 + (long)t * 5;
  const float* wo = w + o * 5;
  float acc = bias[o];
#pragma unroll
  for (int k = 0; k < 5; ++k) acc += xv[k] * wo[k];
  y[idx] = gelu_f(acc);
}

// ---------------------------------------------------------------------------
// Decoder final layer: transposed conv, C=512 -> O=1, k=5 (VALU)
// ---------------------------------------------------------------------------
__global__ void dec_final_k5(const float* __restrict__ x,   // (4,512,16384)
                             const float* __restrict__ w,   // (512,1,5)
                             const float* __restrict__ bias,// (1)
                             float* __restrict__ out)       // (4,81920)
{
  long idx = (long)blockIdx.x * blockDim.x + threadIdx.x;
  if (idx >= (long)4 * 81920) return;
  int to = (int)(idx % 81920);
  int b  = (int)(idx / 81920);
  int t = to / 5, kk = to % 5;
  const float* xb = x + (long)b * 512 * 16384 + t;
  float acc = bias[0];
  for (int c = 0; c < 512; ++c) acc += xb[(long)c * 16384] * w[c * 5 + kk];
  out[idx] = gelu_f(acc);
}

// ---------------------------------------------------------------------------
// RVQ: codebook norms, then 8 sequential stages per (b,t) vector in LDS.
// Deterministic first-min argmin + fixed-order loss reduction.
// ---------------------------------------------------------------------------
__global__ void cb_norms(const float* __restrict__ cb, float* __restrict__ cbn) {
  int idx = blockIdx.x * blockDim.x + threadIdx.x;
  if (idx >= 8 * 1024) return;
  const float* row = cb + (long)idx * 512;
  float s = 0.f;
  for (int c = 0; c < 512; ++c) s += row[c] * row[c];
  cbn[idx] = s;
}

__global__ __launch_bounds__(256) void rvq_kernel(
    const float* __restrict__ xenc, const float* __restrict__ cb,
    const float* __restrict__ cbn, float* __restrict__ xq,
    float* __restrict__ partial)
{
  __shared__ float r[512];
  __shared__ float orig[512];
  __shared__ float redv[256];
  __shared__ int   redj[256];
  __shared__ float s_rr;
  __shared__ int   s_bestj;
  __shared__ float s_loss;

  int vec = blockIdx.x;               // 0..1023 -> (b,t)
  int b = vec >> 8, t = vec & 255;
  int tid = threadIdx.x;              // 256 threads

  for (int c = tid; c < 512; c += 256) {
    float v = xenc[((long)b * 512 + c) * 256 + t];
    r[c] = v; orig[c] = v;
  }
  if (tid == 0) s_loss = 0.f;
  __syncthreads();

  for (int q = 0; q < 8; ++q) {
    // ||r||^2
    redv[tid] = r[tid] * r[tid] + r[tid + 256] * r[tid + 256];
    __syncthreads();
    for (int s = 128; s > 0; s >>= 1) {
      if (tid < s) redv[tid] += redv[tid + s];
      __syncthreads();
    }
    if (tid == 0) s_rr = redv[0];
    __syncthreads();
    float rr = s_rr;

    const float* cbq  = cb  + (long)q * 1024 * 512;
    const float* cbnq = cbn + q * 1024;
    float bestd = INFINITY; int bestj = 0x7fffffff;
    for (int s = 0; s < 4; ++s) {
      int j = s * 256 + tid;
      const float* row = cbq + (long)j * 512;
      float dot = 0.f;
#pragma unroll 4
      for (int c = 0; c < 512; ++c) dot += r[c] * row[c];
      float d = rr - 2.0f * dot + cbnq[j];
      if (d < bestd || (d == bestd && j < bestj)) { bestd = d; bestj = j; }
    }
    redv[tid] = bestd; redj[tid] = bestj;
    __syncthreads();
    for (int s = 128; s > 0; s >>= 1) {
      if (tid < s) {
        float dv = redv[tid + s]; int jv = redj[tid + s];
        if (dv < redv[tid] || (dv == redv[tid] && jv < redj[tid])) {
          redv[tid] = dv; redj[tid] = jv;
        }
      }
      __syncthreads();
    }
    if (tid == 0) { s_bestj = redj[0]; s_loss += redv[0]; }
    __syncthreads();
    const float* sel = cbq + (long)s_bestj * 512;
    r[tid]       -= sel[tid];
    r[tid + 256] -= sel[tid + 256];
    __syncthreads();
  }

  for (int c = tid; c < 512; c += 256)
    xq[(long)vec * 512 + c] = orig[c] - r[c];   // quantized = x - final residual
  if (tid == 0) partial[vec] = s_loss;
}

__global__ void vq_loss_reduce(const float* __restrict__ partial,
                               float* __restrict__ outv) {
  __shared__ float red[256];
  int tid = threadIdx.x;
  float s = 0.f;
  for (int i = tid; i < 1024; i += 256) s += partial[i];
  red[tid] = s;
  __syncthreads();
  for (int st = 128; st > 0; st >>= 1) {
    if (tid < st) red[tid] += red[tid + st];
    __syncthreads();
  }
  if (tid == 0) outv[0] = red[0] / (8.0f * 4.0f * 256.0f * 512.0f);
}

// ---------------------------------------------------------------------------
// h = concat(init, xa[:, :-1]) + xq   (all (B=4, T=256, H=512))
// ---------------------------------------------------------------------------
__global__ void make_h(const float* __restrict__ arout,       // (4,512,256)
                       const float* __restrict__ xq,          // (4,256,512)
                       const float* __restrict__ init_state,  // (512)
                       float* __restrict__ h)                 // (4,256,512)
{
  long idx = (long)blockIdx.x * blockDim.x + threadIdx.x;
  if (idx >= (long)4 * 256 * 512) return;
  int c = (int)(idx & 511);
  int t = (int)((idx >> 9) & 255);
  int b = (int)(idx >> 17);
  float xa = (t == 0) ? init_state[c]
                      : arout[((long)b * 512 + c) * 256 + (t - 1)];
  h[idx] = xa + xq[idx];
}

// ---------------------------------------------------------------------------
// LSTM gates: g = Wcat(2048x1024) @ xh(1024x4) via WMMA (16 cols, 4 valid).
// K loop split in wih/whh halves; paired b64 loads; 32-bit offsets from
// uniform bases. Lanes n>=4 load don't-care data (columns are independent
// in WMMA; their results are masked at the store). All such reads stay
// inside the d_ws allocation by construction, so no guard/EXEC dance needed.
// ---------------------------------------------------------------------------
__global__ __launch_bounds__(256) void lstm_gates(
    const float* __restrict__ wih, const float* __restrict__ whh,
    const float* __restrict__ xsrc, unsigned xstride,
    const float* __restrict__ hsrc,
    float* __restrict__ g)
{
  int wave = (int)(((long)blockIdx.x * blockDim.x + threadIdx.x) >> 5);
  if (wave >= 128) return;
  int lane = threadIdx.x & 31;
  int n  = lane & 15;
  int kb = (lane >> 4) << 1;

  unsigned iaBase = (unsigned)(wave * 16 + n) * 512u + (unsigned)kb;
  unsigned ibx = (unsigned)n * xstride + (unsigned)kb;
  unsigned ibh = (unsigned)n * 512u + (unsigned)kb;

  v8f acc = {};
  // first half: k in [0,512), A from wih, B from x/input
  unsigned ia = iaBase;
#pragma unroll 4
  for (int k = 0; k < 512; k += 4) {
    v2f a = *(const v2f*)(wih + ia);
    v2f b = *(const v2f*)(xsrc + ibx);
    acc = wmma_f32_16x16x4(a, b, acc);
    ia  += 4u;
    ibx += 4u;
  }
  // second half: k in [512,1024), A from whh, B from h state
  ia = iaBase;
#pragma unroll 4
  for (int k = 0; k < 512; k += 4) {
    v2f a = *(const v2f*)(whh + ia);
    v2f b = *(const v2f*)(hsrc + ibh);
    acc = wmma_f32_16x16x4(a, b, acc);
    ia  += 4u;
    ibh += 4u;
  }

  if (n < 4) {
    int rbase = (lane & 16) ? 8 : 0;
#pragma unroll
    for (int r = 0; r < 8; ++r) {
      int mrow = wave * 16 + r + rbase;
      g[n * 2048 + mrow] = acc[r];
    }
  }
}

__global__ void lstm_update(const float* __restrict__ g,
                            const float* __restrict__ bih,
                            const float* __restrict__ bhh,
                            float* __restrict__ h, float* __restrict__ c,
                            float* __restrict__ y, int writeY, int t)
{
  int idx = blockIdx.x * blockDim.x + threadIdx.x;
  if (idx >= 2048) return;
  int n = idx >> 9, j = idx & 511;
  const float* gn = g + n * 2048;
  float gi = gn[j]        + bih[j]        + bhh[j];
  float gf = gn[512 + j]  + bih[512 + j]  + bhh[512 + j];
  float gg = gn[1024 + j] + bih[1024 + j] + bhh[1024 + j];
  float go = gn[1536 + j] + bih[1536 + j] + bhh[1536 + j];
  float cp = c[n * 512 + j];
  float cn = sigm_f(gf) * cp + sigm_f(gi) * tanhf(gg);
  float hn = sigm_f(go) * tanhf(cn);
  c[n * 512 + j] = cn;
  h[n * 512 + j] = hn;
  if (writeY) y[((long)n * 256 + t) * 512 + j] = hn;
}

// ---------------------------------------------------------------------------
// utility kernels
// ---------------------------------------------------------------------------
__global__ void zero_buf(float* p, long nel) {
  long i = (long)blockIdx.x * blockDim.x + threadIdx.x;
  if (i < nel) p[i] = 0.f;
}
__global__ void copy_buf(const float* __restrict__ s, float* __restrict__ d, long nel) {
  long i = (long)blockIdx.x * blockDim.x + threadIdx.x;
  if (i < nel) d[i] = s[i];
}

// ---------------------------------------------------------------------------
// host side
// ---------------------------------------------------------------------------
extern "C" void kernel_launch(void* const* d_in, const int* in_sizes, int n_in,
                              void* d_out, int out_size, void* d_ws, size_t ws_size,
                              hipStream_t stream) {
  (void)in_sizes; (void)n_in; (void)out_size; (void)ws_size;

  const float* wav = (const float*)d_in[0];
  const float *EW[8], *EB[8], *AW[8], *AB[8], *DW[8], *DB[8];
  for (int i = 0; i < 8; ++i) { EW[i] = (const float*)d_in[1 + 2 * i];  EB[i] = (const float*)d_in[2 + 2 * i]; }
  for (int i = 0; i < 8; ++i) { AW[i] = (const float*)d_in[17 + 2 * i]; AB[i] = (const float*)d_in[18 + 2 * i]; }
  for (int i = 0; i < 8; ++i) { DW[i] = (const float*)d_in[33 + 2 * i]; DB[i] = (const float*)d_in[34 + 2 * i]; }
  const float *LWih[2], *LWhh[2], *LBih[2], *LBhh[2];
  for (int l = 0; l < 2; ++l) {
    LWih[l] = (const float*)d_in[49 + 4 * l];
    LWhh[l] = (const float*)d_in[50 + 4 * l];
    LBih[l] = (const float*)d_in[51 + 4 * l];
    LBhh[l] = (const float*)d_in[52 + 4 * l];
  }
  const float* cb         = (const float*)d_in[57];
  const float* init_state = (const float*)d_in[58];

  float* outf = (float*)d_out;                  // [out(327680) | vq(1) | hT(4096) | cT(4096)]

  // workspace layout (floats)
  float* bufA   = (float*)d_ws;                 // 33,554,432  (4*512*16384)
  float* bufB   = bufA   + 33554432L;           // 16,777,216
  float* xenc   = bufB   + 16777216L;           // 524,288  (4,512,256)
  float* xq     = xenc   + 524288L;             // 524,288  (4,256,512)
  float* hsum   = xq     + 524288L;             // 524,288
  float* ylstm  = hsum   + 524288L;             // 524,288  (4,256,512)
  float* arout  = ylstm  + 524288L;             // 524,288  (4,512,256)
  float* gbuf   = arout  + 524288L;             // 8,192    (4 x 2048 gates)
  float* hstate = gbuf   + 8192L;               // 4,096    (2,4,512)
  float* cstate = hstate + 4096L;               // 4,096
  float* partial= cstate + 4096L;               // 1,024
  float* cbn    = partial+ 1024L;               // 8,192
  float* wtbuf  = cbn    + 8192L;               // 524,288  (transposed convT weights)

  auto launch_gemm = [&](bool BP, const float* x, const float* w,
                         const float* bias, float* y,
                         int O, int colT, int R,
                         unsigned wsO, unsigned aKko,
                         unsigned xsB, unsigned xsT,
                         unsigned bStepR, unsigned offB1,
                         unsigned ysB, unsigned ysO, int KOUT) {
    int Mtiles = O / 16;
    int nColTiles = (4 * colT) / 16;
    long waves = (long)Mtiles * nColTiles;
    unsigned blocks = (unsigned)((waves * 32 + 255) / 256);
    dim3 grid(blocks, (unsigned)KOUT, 1);
    if (BP)
      conv_gemm_wmma<true><<<grid, 256, 0, stream>>>(
          x, w, bias, y, Mtiles, nColTiles, colT, R,
          wsO, aKko, xsB, xsT, bStepR, offB1, ysB, ysO, KOUT);
    else
      conv_gemm_wmma<false><<<grid, 256, 0, stream>>>(
          x, w, bias, y, Mtiles, nColTiles, colT, R,
          wsO, aKko, xsB, xsT, bStepR, offB1, ysB, ysO, KOUT);
  };

  // ----- encoder (shared between enc_params and ar_params) -----
  auto encode_net = [&](const float* const* W, const float* const* Bb, float* finalOut) {
    // L0: k=5, C_in=1 -> bufA (4,512,16384)
    {
      long total = (long)4 * 512 * 16384;
      conv0_k5<<<(unsigned)((total + 255) / 256), 256, 0, stream>>>(wav, W[0], Bb[0], bufA, 16384);
    }
    int Tin = 16384;
    const float* cur = bufA;
    for (int i = 1; i <= 6; ++i) {           // k=2 conv layers (WMMA, R=1024)
      int Tout = Tin / 2;
      float* dst = (i & 1) ? bufB : bufA;    // L1->B, L2->A, ... L6->A
      // A idx = o*1024 + r (contiguous); B idx = b*512*Tin + t*2 + r*(Tin/2), pair offs 1
      launch_gemm(true, cur, W[i], Bb[i], dst,
                  512, Tout, 1024,
                  1024u, 0u,
                  (unsigned)(512 * Tin), 2u,
                  (unsigned)(Tin / 2), 1u,
                  (unsigned)(512 * Tout), (unsigned)Tout, 1);
      cur = dst; Tin = Tout;
    }
    // L7: k=1, (B,C,T) layout, T=256 -> finalOut (4,512,256)
    // A idx = o*512 + r; B idx = b*512*256 + t + r*256 (pair offs 256)
    launch_gemm(false, cur, W[7], Bb[7], finalOut,
                512, 256, 512,
                512u, 0u,
                (unsigned)(512 * 256), 1u,
                256u, 256u,
                (unsigned)(512 * 256), 256u, 1);
  };

  // 1) main encoder -> xenc
  encode_net(EW, EB, xenc);

  // 2) RVQ -> xq, vq_loss
  cb_norms<<<(8 * 1024 + 255) / 256, 256, 0, stream>>>(cb, cbn);
  rvq_kernel<<<1024, 256, 0, stream>>>(xenc, cb, cbn, xq, partial);
  vq_loss_reduce<<<1, 256, 0, stream>>>(partial, outf + 327680);

  // 3) AR encoder -> arout (bufA/bufB free again)
  encode_net(AW, AB, arout);

  // 4) h = shifted(xa) + xq
  make_h<<<(unsigned)((524288L + 255) / 256), 256, 0, stream>>>(arout, xq, init_state, hsum);

  // 5) LSTM: 256 steps x 2 layers; gates via WMMA, state in ws
  zero_buf<<<(8192 + 255) / 256, 256, 0, stream>>>(hstate, 8192);   // h + c contiguous
  for (int t = 0; t < 256; ++t) {
    for (int l = 0; l < 2; ++l) {
      const float* xsrc; unsigned xstride;
      if (l == 0) { xsrc = hsum + (long)t * 512; xstride = 256u * 512u; }
      else        { xsrc = hstate;               xstride = 512u; }
      const float* hsrc = hstate + (long)l * 2048;
      lstm_gates<<<16, 256, 0, stream>>>(LWih[l], LWhh[l], xsrc, xstride, hsrc, gbuf);
      lstm_update<<<8, 256, 0, stream>>>(gbuf, LBih[l], LBhh[l],
                                         hstate + (long)l * 2048,
                                         cstate + (long)l * 2048,
                                         ylstm, (l == 1) ? 1 : 0, t);
    }
  }
  copy_buf<<<(4096 + 255) / 256, 256, 0, stream>>>(hstate, outf + 327681, 4096);
  copy_buf<<<(4096 + 255) / 256, 256, 0, stream>>>(cstate, outf + 331777, 4096);

  // 6) decoder
  // L0: k=1 conv reading ylstm in (B,T,H) layout -> bufA (4,512,256)
  // A idx = o*512 + r; B idx = b*256*512 + t*512 + r (contiguous pairs)
  launch_gemm(true, ylstm, DW[0], DB[0], bufA,
              512, 256, 512,
              512u, 0u,
              (unsigned)(256 * 512), 512u,
              1u, 1u,
              (unsigned)(512 * 256), 256u, 1);
  {
    int Tin = 256;
    const float* cur = bufA;
    for (int i = 1; i <= 6; ++i) {           // transposed conv k=2 (WMMA, R=512, grid.y=2)
      int Tout = Tin * 2;
      float* dst = (i & 1) ? bufB : bufA;    // L1->B, ..., L6->A (134MB in A, fits)
      // transpose weights (C,O,2) -> (2,O,C) so A pairs are contiguous
      transpose_w_cok<<<(512 * 512 * 2 + 255) / 256, 256, 0, stream>>>(DW[i], wtbuf);
      // A idx = kko*262144 + o*512 + r; B idx = b*512*Tin + t + r*Tin (pair offs Tin)
      launch_gemm(false, cur, wtbuf, DB[i], dst,
                  512, Tin, 512,
                  512u, 262144u,
                  (unsigned)(512 * Tin), 1u,
                  (unsigned)Tin, (unsigned)Tin,
                  (unsigned)(512 * Tout), (unsigned)Tout, 2);
      cur = dst; Tin = Tout;
    }
    // final transposed conv k=5, O=1 -> out
    dec_final_k5<<<(unsigned)(((long)4 * 81920 + 255) / 256), 256, 0, stream>>>(
        cur, DW[7], DB[7], outf);
  }
}